// CompGCNLayer_46454366273980
// MI455X (gfx1250) — compile-verified
//
#include <hip/hip_runtime.h>

typedef __attribute__((ext_vector_type(2))) float v2f;
typedef __attribute__((ext_vector_type(8))) float v8f;

#define N_NODES 100000
#define E2_TOT  1200000
#define E_HALF  600000
#define NREL    1000        // 2*R rows of rel_embed (loop row never indexed by edges)
#define LN_EPS  1e-5f
#define AST     388         // LDS stride for 16x384 A tile (388 % 64 == 4 -> conflict-free col reads)
#define RST     132         // LDS stride for 16x128 rel tile
#define TPB     5           // row tiles per block in the fused GEMM (6250 = 1250 * 5)

// ---------------- degree count ----------------
__global__ void k_deg(const int* __restrict__ ei, int* deg_in, int* deg_out) {
    int e = blockIdx.x * blockDim.x + threadIdx.x;
    if (e >= E2_TOT) return;
    int src = ei[e];                       // row 0 of edge_index
    atomicAdd((e < E_HALF ? deg_in : deg_out) + src, 1);
}

__global__ void k_inv(const int* __restrict__ deg_in, const int* __restrict__ deg_out,
                      float* inv_in, float* inv_out) {
    int n = blockIdx.x * blockDim.x + threadIdx.x;
    if (n >= N_NODES) return;
    int d0 = deg_in[n];
    int d1 = deg_out[n];
    inv_in[n]  = d0 > 0 ? 1.0f / (float)d0 : 0.0f;
    inv_out[n] = d1 > 0 ? 1.0f / (float)d1 : 0.0f;
}

// ---------------- fold loop_rel into W_loop:  Wl2 = diag(loop_rel) @ Wloop ----------------
__global__ void k_wloop(const float* __restrict__ Wloop, const float* __restrict__ loop_rel,
                        float* __restrict__ Wl2) {
    int i = blockIdx.x * blockDim.x + threadIdx.x;
    if (i >= 128 * 128) return;
    Wl2[i] = Wloop[i] * loop_rel[i >> 7];
}

// ------- edge scatter: A[src] += inv_deg[src] * x[dst] * rel[t]  (one wave32 per edge) -------
__global__ void __launch_bounds__(256) k_scatter(const float* __restrict__ x,
                                                 const int* __restrict__ ei,
                                                 const int* __restrict__ et,
                                                 const float* __restrict__ rel,
                                                 const float* __restrict__ inv_in,
                                                 const float* __restrict__ inv_out,
                                                 float* A_in, float* A_out) {
    int lane = threadIdx.x & 31;
    int e = blockIdx.x * 8 + (threadIdx.x >> 5);
    if (e >= E2_TOT) return;
    int src = ei[e];
    int dst = ei[E2_TOT + e];              // row 1 of edge_index
    int t   = et[e];
    float nrm = (e < E_HALF ? inv_in : inv_out)[src];
    float4 xd = ((const float4*)(x   + (size_t)dst * 128))[lane];
    float4 rl = ((const float4*)(rel + (size_t)t   * 128))[lane];
    float* acc = (e < E_HALF ? A_in : A_out) + (size_t)src * 128 + lane * 4;
    atomicAdd(acc + 0, nrm * xd.x * rl.x);
    atomicAdd(acc + 1, nrm * xd.y * rl.y);
    atomicAdd(acc + 2, nrm * xd.z * rl.z);
    atomicAdd(acc + 3, nrm * xd.w * rl.w);
}

// ---- fused GEMM: [A_in | A_out | x] (100000x384) @ [Win; Wout; Wl2] (384x128), async-staged ----
__global__ void __launch_bounds__(256) k_gemm(const float* __restrict__ A_in,
                                              const float* __restrict__ A_out,
                                              const float* __restrict__ x,
                                              const float* __restrict__ Win,
                                              const float* __restrict__ Wout,
                                              const float* __restrict__ Wl2,
                                              const float* __restrict__ bias,
                                              float* __restrict__ out_pre) {
    __shared__ __align__(16) float As[2][16 * AST];
    const int tid = threadIdx.x;
    const int tile0 = blockIdx.x * TPB;

    const float* seg[3] = {A_in, A_out, x};

    // Async-stage row tile t into ping-pong buffer b: pure copy, 16B chunks, direct to LDS.
    // 16 rows x 384 floats; thread (r = tid>>4) handles 6 b128 chunks of its row.
    auto stage = [&](int t, int b) {
        const int r  = tid >> 4;
        const int c0 = (tid & 15) * 4;                       // starting float col of chunk
        const size_t rowoff = ((size_t)t * 16 + r) * 128;    // per-segment row offset (floats)
        const unsigned lrow = (unsigned)(size_t)(&As[b][0]) + (unsigned)(r * AST * 4);
#pragma unroll
        for (int j = 0; j < 6; ++j) {
            const int kc = c0 + j * 64;                      // col within 0..383
            const int s  = kc >> 7;                          // segment select (const after unroll)
            const int k  = kc & 127;
            unsigned goff = (unsigned)((rowoff + k) * 4);    // byte offset from segment base
            unsigned loff = lrow + (unsigned)kc * 4;         // LDS byte address
            asm volatile("global_load_async_to_lds_b128 %0, %1, %2"
                         :: "v"(loff), "v"(goff), "s"(seg[s]) : "memory");
        }
    };

    stage(tile0, 0);
    asm volatile("s_wait_asynccnt 0x0" ::: "memory");
    __syncthreads();

    const int lane  = tid & 31;
    const int nbase = (tid >> 5) * 16;     // 8 waves -> 8 column tiles of 16
    const int m     = lane & 15;
    const int kh    = (lane >> 4) * 2;     // lanes 0-15 -> K{0,1}, lanes 16-31 -> K{2,3}
    const float* Ws[3] = {Win, Wout, Wl2};
    const float bcol = bias[nbase + m];

    for (int it = 0; it < TPB; ++it) {
        if (it + 1 < TPB) stage(tile0 + it + 1, (it + 1) & 1);   // prefetch next tile (async)

        const float* Ab = &As[it & 1][0];
        v8f acc = {0.f, 0.f, 0.f, 0.f, 0.f, 0.f, 0.f, 0.f};
#pragma unroll
        for (int w = 0; w < 3; ++w) {
            const float* W = Ws[w];
            const int koff = w * 128;
#pragma unroll 8
            for (int kt = 0; kt < 32; ++kt) {
                int k0 = kt * 4 + kh;
                v2f a, b;
                a.x = Ab[m * AST + koff + k0];
                a.y = Ab[m * AST + koff + k0 + 1];
                b.x = W[(size_t)k0 * 128 + nbase + m];
                b.y = W[(size_t)(k0 + 1) * 128 + nbase + m];
                acc = __builtin_amdgcn_wmma_f32_16x16x4_f32(false, a, false, b,
                                                            (short)0, acc, false, false);
            }
        }

        const size_t rowbase = (size_t)(tile0 + it) * 16;
#pragma unroll
        for (int i = 0; i < 8; ++i) {
            int M = (lane >> 4) * 8 + i;   // C/D layout: VGPR i = row i (lanes 0-15) / row 8+i
            out_pre[(rowbase + M) * 128 + nbase + m] = acc[i] * (1.0f / 3.0f) + bcol;
        }

        asm volatile("s_wait_asynccnt 0x0" ::: "memory");  // my prefetch done
        __syncthreads();                                    // everyone's prefetch done / reads done
    }
}

// ---------------- per-column (axis=0) statistics ----------------
#define ROWS_PER_BLOCK 200
__global__ void __launch_bounds__(256) k_colstats(const float* __restrict__ out_pre,
                                                  float* colsum, float* colsumsq) {
    __shared__ float s1[256], s2[256];
    int c = threadIdx.x & 127;
    int h = threadIdx.x >> 7;
    size_t start = (size_t)blockIdx.x * ROWS_PER_BLOCK;
    size_t end = start + ROWS_PER_BLOCK;
    if (end > N_NODES) end = N_NODES;
    float s = 0.f, q = 0.f;
    for (size_t r = start + h; r < end; r += 2) {
        float v = out_pre[r * 128 + c];
        s += v; q += v * v;
    }
    s1[threadIdx.x] = s; s2[threadIdx.x] = q;
    __syncthreads();
    if (h == 0) {
        s += s1[threadIdx.x + 128];
        q += s2[threadIdx.x + 128];
        atomicAdd(&colsum[c], s);
        atomicAdd(&colsumsq[c], q);
    }
}

__global__ void k_finstats(const float* __restrict__ colsum, const float* __restrict__ colsumsq,
                           float* mean, float* rstd) {
    int c = threadIdx.x;
    float m = colsum[c] * (1.0f / N_NODES);
    float v = colsumsq[c] * (1.0f / N_NODES) - m * m;
    mean[c] = m;
    rstd[c] = rsqrtf(v + LN_EPS);
}

__global__ void __launch_bounds__(256) k_norm(float* __restrict__ out,
                                              const float* __restrict__ mean,
                                              const float* __restrict__ rstd,
                                              const float* __restrict__ gamma,
                                              const float* __restrict__ beta) {
    size_t i = (size_t)blockIdx.x * 256 + threadIdx.x;
    if (i >= (size_t)N_NODES * 128) return;
    int c = (int)(i & 127);
    out[i] = (out[i] - mean[c]) * rstd[c] * gamma[c] + beta[c];
}

// ---------------- rel_out = rel_embed (1000x128) @ Wrel ----------------
__global__ void __launch_bounds__(256) k_relgemm(const float* __restrict__ rel,
                                                 const float* __restrict__ Wrel,
                                                 float* __restrict__ rel_out) {
    __shared__ float Rs[16 * RST];
    const int tid = threadIdx.x;
    const int rowbase = blockIdx.x * 16;   // 63 tiles, last one partial
    for (int idx = tid; idx < 16 * 128; idx += 256) {
        int r = idx >> 7, k = idx & 127;
        int row = rowbase + r;
        Rs[r * RST + k] = (row < NREL) ? rel[(size_t)row * 128 + k] : 0.0f;
    }
    __syncthreads();

    const int lane  = tid & 31;
    const int nbase = (tid >> 5) * 16;
    const int m     = lane & 15;
    const int kh    = (lane >> 4) * 2;
    v8f acc = {0.f, 0.f, 0.f, 0.f, 0.f, 0.f, 0.f, 0.f};
#pragma unroll 8
    for (int kt = 0; kt < 32; ++kt) {
        int k0 = kt * 4 + kh;
        v2f a, b;
        a.x = Rs[m * RST + k0];
        a.y = Rs[m * RST + k0 + 1];
        b.x = Wrel[(size_t)k0 * 128 + nbase + m];
        b.y = Wrel[(size_t)(k0 + 1) * 128 + nbase + m];
        acc = __builtin_amdgcn_wmma_f32_16x16x4_f32(false, a, false, b,
                                                    (short)0, acc, false, false);
    }
#pragma unroll
    for (int i = 0; i < 8; ++i) {
        int row = rowbase + (lane >> 4) * 8 + i;
        if (row < NREL) rel_out[(size_t)row * 128 + nbase + m] = acc[i];
    }
}

extern "C" void kernel_launch(void* const* d_in, const int* in_sizes, int n_in,
                              void* d_out, int out_size, void* d_ws, size_t ws_size,
                              hipStream_t stream) {
    const float* x        = (const float*)d_in[0];
    const int*   ei       = (const int*)d_in[1];
    const int*   et       = (const int*)d_in[2];
    const float* rel      = (const float*)d_in[3];
    const float* Win      = (const float*)d_in[4];
    const float* Wout     = (const float*)d_in[5];
    const float* Wrel     = (const float*)d_in[6];
    const float* Wloop    = (const float*)d_in[7];
    const float* loop_rel = (const float*)d_in[8];
    const float* bias     = (const float*)d_in[9];
    const float* gamma    = (const float*)d_in[10];
    const float* beta     = (const float*)d_in[11];

    float* out     = (float*)d_out;                       // 100000x128, also used as out_pre
    float* rel_out = out + (size_t)N_NODES * 128;         // 1000x128

    char* ws = (char*)d_ws;
    size_t off = 0;
    float* A_in  = (float*)(ws + off); off += (size_t)N_NODES * 128 * 4;
    float* A_out = (float*)(ws + off); off += (size_t)N_NODES * 128 * 4;
    int* deg_in  = (int*)(ws + off);   off += (size_t)N_NODES * 4;
    int* deg_out = (int*)(ws + off);   off += (size_t)N_NODES * 4;
    float* colsum   = (float*)(ws + off); off += 128 * 4;
    float* colsumsq = (float*)(ws + off); off += 128 * 4;
    size_t zero_bytes = off;                              // everything above must start at 0
    float* inv_in  = (float*)(ws + off); off += (size_t)N_NODES * 4;
    float* inv_out = (float*)(ws + off); off += (size_t)N_NODES * 4;
    float* mean    = (float*)(ws + off); off += 512;
    float* rstd    = (float*)(ws + off); off += 512;
    float* Wl2     = (float*)(ws + off); off += 128 * 128 * 4;

    hipMemsetAsync(d_ws, 0, zero_bytes, stream);

    k_deg<<<(E2_TOT + 255) / 256, 256, 0, stream>>>(ei, deg_in, deg_out);
    k_inv<<<(N_NODES + 255) / 256, 256, 0, stream>>>(deg_in, deg_out, inv_in, inv_out);
    k_wloop<<<(128 * 128 + 255) / 256, 256, 0, stream>>>(Wloop, loop_rel, Wl2);
    k_scatter<<<E2_TOT / 8, 256, 0, stream>>>(x, ei, et, rel, inv_in, inv_out, A_in, A_out);
    k_gemm<<<N_NODES / (16 * TPB), 256, 0, stream>>>(A_in, A_out, x, Win, Wout, Wl2, bias, out);
    k_colstats<<<(N_NODES + ROWS_PER_BLOCK - 1) / ROWS_PER_BLOCK, 256, 0, stream>>>(out, colsum, colsumsq);
    k_finstats<<<1, 128, 0, stream>>>(colsum, colsumsq, mean, rstd);
    k_norm<<<(int)(((size_t)N_NODES * 128 + 255) / 256), 256, 0, stream>>>(out, mean, rstd, gamma, beta);
    k_relgemm<<<(NREL + 15) / 16, 256, 0, stream>>>(rel, Wrel, rel_out);
}